// GCN_Decoder_29738353557974
// MI455X (gfx1250) — compile-verified
//
#include <hip/hip_runtime.h>
#include <hip/hip_bf16.h>

// Problem constants (match reference)
#define DNODES 100000
#define DDIM   128
#define DEDGE  600000
#define BN_EPS 1e-5f

typedef __attribute__((ext_vector_type(2))) float v2f;
typedef __attribute__((ext_vector_type(8))) float v8f;

// ---------------------------------------------------------------------------
// Small utility kernels
// ---------------------------------------------------------------------------
__global__ void fill_kernel(float* __restrict__ p, float val, int n) {
    int i = blockIdx.x * blockDim.x + threadIdx.x;
    if (i < n) p[i] = val;
}

// out/in degree accumulation for all three relations in one launch.
__global__ void degree_kernel(const int* __restrict__ s0, const int* __restrict__ d0,
                              const int* __restrict__ s1, const int* __restrict__ d1,
                              const int* __restrict__ s2, const int* __restrict__ d2,
                              float* __restrict__ ns0, float* __restrict__ nd0,
                              float* __restrict__ ns1, float* __restrict__ nd1,
                              float* __restrict__ ns2, float* __restrict__ nd2,
                              int ne) {
    int i = blockIdx.x * blockDim.x + threadIdx.x;
    if (i >= 3 * ne) return;
    int r = i / ne;
    int e = i - r * ne;
    const int* s = (r == 0) ? s0 : (r == 1) ? s1 : s2;
    const int* d = (r == 0) ? d0 : (r == 1) ? d1 : d2;
    float* ns = (r == 0) ? ns0 : (r == 1) ? ns1 : ns2;
    float* nd = (r == 0) ? nd0 : (r == 1) ? nd1 : nd2;
    unsafeAtomicAdd(&ns[s[e]], 1.0f);   // native global_atomic_add_f32
    unsafeAtomicAdd(&nd[d[e]], 1.0f);
}

// deg -> clamp(deg,1)^-0.5 in place (runs over all 6 contiguous norm arrays)
__global__ void norm_kernel(float* __restrict__ p, int n) {
    int i = blockIdx.x * blockDim.x + threadIdx.x;
    if (i < n) p[i] = rsqrtf(fmaxf(p[i], 1.0f));
}

// agg[v, c] = brel[l,0,c] + brel[l,1,c] + brel[l,2,c]
__global__ void init_agg_kernel(float* __restrict__ agg,
                                const float* __restrict__ brel_l, int total) {
    int i = blockIdx.x * blockDim.x + threadIdx.x;
    if (i >= total) return;
    int c = i & (DDIM - 1);
    agg[i] = brel_l[c] + brel_l[DDIM + c] + brel_l[2 * DDIM + c];
}

// ---------------------------------------------------------------------------
// Edge scatter: agg[d] += tmp[s] * nd[d]   (one wave32 per edge, float4 lanes)
// ---------------------------------------------------------------------------
__global__ __launch_bounds__(256) void scatter_kernel(
        const float* __restrict__ tmp, const int* __restrict__ src,
        const int* __restrict__ dst, const float* __restrict__ nd,
        float* __restrict__ agg, int ne) {
    int e = blockIdx.x * 8 + (threadIdx.x >> 5);
    if (e >= ne) return;
    int lane = threadIdx.x & 31;
    int s = src[e];              // scalar (wave-uniform) loads
    int d = dst[e];
    float w = nd[d];
    const float4* p = (const float4*)(tmp + (size_t)s * DDIM) + lane;
    float4 v = *p;
    float* q = agg + (size_t)d * DDIM + lane * 4;
    unsafeAtomicAdd(q + 0, v.x * w);
    unsafeAtomicAdd(q + 1, v.y * w);
    unsafeAtomicAdd(q + 2, v.z * w);
    unsafeAtomicAdd(q + 3, v.w * w);
}

// ---------------------------------------------------------------------------
// WMMA f32 GEMM: out[N x 128] = A[N x 128] * W[128 x 128]  (+ epilogue)
// 256 threads = 8 wave32s; each wave computes a 16x128 strip using
// V_WMMA_F32_16X16X4_F32 (exact fp32 math, matches reference precision).
//
// W is staged in LDS with row pairs interleaved:
//   element (k, c) lives at ((k>>1)*128 + c')*2 + (k&1),  c' = c ^ ((pair&1)<<4)
// so each B-fragment (W[kk][c], W[kk+1][c]) is ONE aligned ds_load_b64
// directly into the even-aligned WMMA source pair (no register shuffles),
// and the pair-parity XOR swizzle puts the two wave halves (pair even vs
// odd) in disjoint 32-bank groups -> conflict-free b64 loads.
// ---------------------------------------------------------------------------
template <bool ROWSCALE, bool COLBIAS, bool BN>
__global__ __launch_bounds__(256) void gemm128_kernel(
        const float* __restrict__ A, const float* __restrict__ W,
        float* __restrict__ out,
        const float* __restrict__ rowscale,   // ns (per output row) or null
        const float* __restrict__ colbias,    // bfc (per column) or null
        const float* __restrict__ bn_g, const float* __restrict__ bn_b,
        const float* __restrict__ bn_mu, const float* __restrict__ bn_var,
        int nstrips) {
    __shared__ float Wl[DDIM * DDIM];

    // Cooperative load of W (row-major) into interleaved-pair LDS layout.
    const float4* Wv = (const float4*)W;
    for (int i = threadIdx.x; i < (DDIM * DDIM) / 4; i += 256) {
        int row  = i >> 5;               // 32 float4s per row
        int c0   = (i & 31) * 4;
        float4 v = Wv[i];
        int pair = row >> 1;
        int lo   = row & 1;
        int cx   = c0 ^ ((pair & 1) << 4);     // bank-group swizzle
        int base = (pair * DDIM + cx) * 2 + lo;
        Wl[base + 0] = v.x;
        Wl[base + 2] = v.y;
        Wl[base + 4] = v.z;
        Wl[base + 6] = v.w;
    }
    __syncthreads();

    int wave = threadIdx.x >> 5;
    int lane = threadIdx.x & 31;
    int strip = blockIdx.x * 8 + wave;
    if (strip >= nstrips) return;       // wave-uniform: EXEC stays all-ones

    int half = lane >> 4;               // 0: lanes 0-15, 1: lanes 16-31
    int l16  = lane & 15;
    const float* Arow = A + (size_t)(strip * 16 + l16) * DDIM;

    // Per-lane B base: this lane reads pair = k/2 + half, column l16 of each
    // tile; the XOR swizzle folds into compile-time tile offsets tofs[t].
    const float* Bp = &Wl[half * 256 + l16 * 2];
    int tofs[8];
#pragma unroll
    for (int t = 0; t < 8; ++t) tofs[t] = ((t * 16) ^ (half << 4)) * 2;

    v8f acc[8] = {};                    // 16x128 strip: 8 N-tiles of 16x16

    for (int k = 0; k < DDIM; k += 4) {
        v2f a = *(const v2f*)(Arow + k + half * 2);   // A frag: K=kk, kk+1
        const float* Bk = Bp + (k >> 1) * 256;
#pragma unroll
        for (int t = 0; t < 8; ++t) {
            v2f b = *(const v2f*)(Bk + tofs[t]);      // one ds_load_b64
            acc[t] = __builtin_amdgcn_wmma_f32_16x16x4_f32(
                false, a, false, b, (short)0, acc[t], false, false);
        }
    }

    // Per-column epilogue constants (column depends only on tile t and l16).
    float colb[8], bs[8], bsh[8];
#pragma unroll
    for (int t = 0; t < 8; ++t) {
        int c = t * 16 + l16;
        if (COLBIAS) colb[t] = colbias[c];
        if (BN) {
            float s = bn_g[c] * rsqrtf(bn_var[c] + BN_EPS);
            bs[t]  = s;
            bsh[t] = bn_b[c] - bn_mu[c] * s;
        }
    }

#pragma unroll
    for (int j = 0; j < 8; ++j) {
        int m   = j + half * 8;          // C/D layout: VGPR j -> M=j (+8 hi half)
        int row = strip * 16 + m;
        float rs = ROWSCALE ? rowscale[row] : 1.0f;
        float* orow = out + (size_t)row * DDIM + l16;
#pragma unroll
        for (int t = 0; t < 8; ++t) {
            float v = acc[t][j];
            if (ROWSCALE) v *= rs;
            if (COLBIAS)  v += colb[t];
            if (BN) { v = v * bs[t] + bsh[t]; v = fmaxf(v, 0.0f); }
            orow[t * 16] = v;
        }
    }
}

// ---------------------------------------------------------------------------
// Host-side launcher
// ---------------------------------------------------------------------------
extern "C" void kernel_launch(void* const* d_in, const int* in_sizes, int n_in,
                              void* d_out, int out_size, void* d_ws, size_t ws_size,
                              hipStream_t stream) {
    const float* x    = (const float*)d_in[0];
    const int*   src[3] = { (const int*)d_in[1], (const int*)d_in[3], (const int*)d_in[5] };
    const int*   dst[3] = { (const int*)d_in[2], (const int*)d_in[4], (const int*)d_in[6] };
    const float* Wrel = (const float*)d_in[7];   // [3][3][128][128]
    const float* brel = (const float*)d_in[8];   // [3][3][128]
    const float* Wfc  = (const float*)d_in[9];   // [3][128][128]
    const float* bfc  = (const float*)d_in[10];  // [3][128]
    const float* gamma = (const float*)d_in[11]; // [2][128]
    const float* beta  = (const float*)d_in[12];
    const float* rmean = (const float*)d_in[13];
    const float* rvar  = (const float*)d_in[14];

    const int N = in_sizes[0] / DDIM;    // 100000
    const int E = in_sizes[1];           // 600000
    const int ND = N * DDIM;
    const int nstrips = N / 16;          // 6250 (N is a multiple of 16)

    // Workspace carve-up (floats)
    float* ws   = (float*)d_ws;
    float* hA   = ws;                    // layer-0 output
    float* hB   = hA + (size_t)ND;       // layer-1 output
    float* tmp  = hB + (size_t)ND;       // per-relation GEMM result
    float* agg  = tmp + (size_t)ND;      // accumulated messages
    float* norm = agg + (size_t)ND;      // 6 * N: ns0,nd0,ns1,nd1,ns2,nd2
    float* ns[3] = { norm,             norm + 2 * (size_t)N, norm + 4 * (size_t)N };
    float* nd[3] = { norm + (size_t)N, norm + 3 * (size_t)N, norm + 5 * (size_t)N };

    const int T = 256;
    // 1) degree norms (recomputed every call: deterministic, no cached state)
    fill_kernel<<<(6 * N + T - 1) / T, T, 0, stream>>>(norm, 0.0f, 6 * N);
    degree_kernel<<<(3 * E + T - 1) / T, T, 0, stream>>>(
        src[0], dst[0], src[1], dst[1], src[2], dst[2],
        ns[0], nd[0], ns[1], nd[1], ns[2], nd[2], E);
    norm_kernel<<<(6 * N + T - 1) / T, T, 0, stream>>>(norm, 6 * N);

    const int gemm_blocks = (nstrips + 7) / 8;
    const int scat_blocks = (E + 7) / 8;

    for (int l = 0; l < 3; ++l) {
        const float* hin  = (l == 0) ? x  : (l == 1) ? hA : hB;
        float*       hout = (l == 0) ? hA : (l == 1) ? hB : (float*)d_out;

        // agg <- sum_r brel[l,r] (broadcast rows)
        init_agg_kernel<<<(ND + T - 1) / T, T, 0, stream>>>(
            agg, brel + (size_t)l * 3 * DDIM, ND);

        for (int r = 0; r < 3; ++r) {
            // tmp = (hin @ Wrel[l,r]) * ns_r[:,None]
            gemm128_kernel<true, false, false><<<gemm_blocks, T, 0, stream>>>(
                hin, Wrel + ((size_t)l * 3 + r) * DDIM * DDIM, tmp,
                ns[r], nullptr, nullptr, nullptr, nullptr, nullptr, nstrips);
            // agg[d] += tmp[s] * nd_r[d]
            scatter_kernel<<<scat_blocks, T, 0, stream>>>(
                tmp, src[r], dst[r], nd[r], agg, E);
        }

        if (l < 2) {
            gemm128_kernel<false, true, true><<<gemm_blocks, T, 0, stream>>>(
                agg, Wfc + (size_t)l * DDIM * DDIM, hout,
                nullptr, bfc + (size_t)l * DDIM,
                gamma + (size_t)l * DDIM, beta + (size_t)l * DDIM,
                rmean + (size_t)l * DDIM, rvar + (size_t)l * DDIM, nstrips);
        } else {
            gemm128_kernel<false, true, false><<<gemm_blocks, T, 0, stream>>>(
                agg, Wfc + (size_t)l * DDIM * DDIM, hout,
                nullptr, bfc + (size_t)l * DDIM,
                nullptr, nullptr, nullptr, nullptr, nstrips);
        }
    }
}